// MultiHeadAdditiveAttentionLayer_3040836845992
// MI455X (gfx1250) — compile-verified
//
#include <hip/hip_runtime.h>
#include <hip/hip_bf16.h>

// ---------------------------------------------------------------------------
// MultiHeadAdditiveAttention on MI455X (gfx1250, wave32)
//   B=4, L=512, HID=512, H=8, d=64
//   outputs: out [4,512,512] then attn [4,8,512,512] concatenated in d_out
// GEMMs use V_WMMA_F32_16X16X4_F32. LDS staging in the attention kernel uses
// gfx1250 GLOBAL_LOAD_ASYNC_TO_LDS_B128 (ASYNCcnt) so the V tile copy overlaps
// the softmax phase. The tanh score phase (537M trans ops) dominates and runs
// as a tight LDS-resident VALU loop.
// ---------------------------------------------------------------------------

typedef float v2f __attribute__((ext_vector_type(2)));
typedef float v8f __attribute__((ext_vector_type(8)));

#define N_HEADS_ 8
#define HEAD_DIM_ 64
#define B_ 4
#define L_ 512
#define HID_ 512
#define QT_ 32        // q rows per workgroup in attn kernel
#define SCP_ 516      // scS row pitch (floats): 8B-aligned rows, bank-staggered
#define KPP_ 65       // KpS/VS row pitch (floats): bank = (k+d)%64

// tanh(x) = 1 - 2/(exp(2x)+1); clamp to avoid inf/inf
__device__ __forceinline__ float fast_tanh(float x) {
    x = fminf(fmaxf(x, -12.0f), 12.0f);
    float e = __expf(2.0f * x);
    return 1.0f - 2.0f * __frcp_rn(e + 1.0f);
}

// gfx1250 async copy: 16 bytes global -> LDS, no VGPR round trip (ASYNCcnt).
// Inline asm (not the builtin) so it works on both ROCm7.2 and amdgpu-toolchain.
__device__ __forceinline__ void async_copy_b128(uint32_t lds_off, const float* gsrc) {
    uint64_t ga = (uint64_t)(uintptr_t)gsrc;
    asm volatile("global_load_async_to_lds_b128 %0, %1, off"
                 :: "v"(lds_off), "v"(ga) : "memory");
}
__device__ __forceinline__ void wait_async_all() {
    asm volatile("s_wait_asynccnt 0x0" ::: "memory");
}

// ---------------------------------------------------------------------------
// Kernel 1: head projections.  z=0: Qp = Q*W2^T, z=1: Kp = K*W1^T
// One wave per 16x16 output tile of a [512,64]x[64,64] per-head GEMM.
// ---------------------------------------------------------------------------
__global__ __launch_bounds__(256) void proj_kernel(
    const float* __restrict__ q_in, const float* __restrict__ k_in,
    const float* __restrict__ W2,   const float* __restrict__ W1,
    float* __restrict__ Qp, float* __restrict__ Kp)
{
    const int which = blockIdx.z;               // 0 -> Qp, 1 -> Kp
    const float* in = which ? k_in : q_in;
    const float* W  = which ? W1   : W2;
    float* outp     = which ? Kp   : Qp;

    const int bh = blockIdx.y;                  // 0..31
    const int b  = bh >> 3, h = bh & 7;

    const int lane = threadIdx.x & 31;
    const int wave = threadIdx.x >> 5;
    const int tile = blockIdx.x * 8 + wave;     // 0..127
    const int mt = tile & 31;                   // 32 row tiles (L=512)
    const int nt = tile >> 5;                   // 4 col tiles (d=64)
    const int l0 = mt * 16, e0 = nt * 16;

    const int hi = lane >> 4;                   // 0/1 half of wave
    const int ln = lane & 15;

    // A[m][k]: row m = l0+ln of input head slice; B[k][n] = W[e0+ln][k]
    const float* arow = in + ((size_t)(b * L_ + l0 + ln)) * HID_ + h * HEAD_DIM_;
    const float* brow = W + (size_t)(e0 + ln) * HEAD_DIM_;

    v8f c = {};
#pragma unroll
    for (int kk = 0; kk < HEAD_DIM_; kk += 4) {
        v2f a, bb;
        a.x  = arow[kk + 2 * hi];
        a.y  = arow[kk + 2 * hi + 1];
        bb.x = brow[kk + 2 * hi];
        bb.y = brow[kk + 2 * hi + 1];
        c = __builtin_amdgcn_wmma_f32_16x16x4_f32(false, a, false, bb,
                                                  (short)0, c, false, false);
    }

    float* orow = outp + (size_t)bh * (L_ * HEAD_DIM_);
#pragma unroll
    for (int r = 0; r < 8; ++r) {
        int m = l0 + r + 8 * hi;
        orow[(size_t)m * HEAD_DIM_ + e0 + ln] = c[r];
    }
}

// ---------------------------------------------------------------------------
// Kernel 2: additive scores + softmax + attn@V.
// One workgroup (256 thr = 8 waves) per (b, h, 32 q-rows).
// LDS: Kp[512][65] (reused for V), scores[32][516], Qp[32][65], v[64] ~208KB.
// ---------------------------------------------------------------------------
__global__ __launch_bounds__(256) void attn_kernel(
    const float* __restrict__ Qp, const float* __restrict__ Kp,
    const float* __restrict__ value, const float* __restrict__ vvec,
    float* __restrict__ attn_out, float* __restrict__ x_out)
{
    extern __shared__ float smem[];
    float* KpS = smem;                      // 512*65  (reused for V later)
    float* scS = KpS + L_ * KPP_;           // 32*516
    float* QpS = scS + QT_ * SCP_;          // 32*65
    float* vS  = QpS + QT_ * KPP_;          // 64

    const int bh = blockIdx.y;
    const int b = bh >> 3, h = bh & 7;
    const int q0 = blockIdx.x * QT_;
    const int tid = threadIdx.x;
    const int lane = tid & 31, wave = tid >> 5;

    // LDS byte offsets for the async-copy destinations
    const uint32_t ldsKp = (uint32_t)(uintptr_t)KpS;
    const uint32_t ldsQp = (uint32_t)(uintptr_t)QpS;

    // ---- async-stage Kp (whole head) and Qp tile into LDS ----
    const float* KpG = Kp + (size_t)bh * (L_ * HEAD_DIM_);
    for (int i = tid; i < L_ * (HEAD_DIM_ / 4); i += 256) {
        int k = i >> 4, d4 = (i & 15) * 4;
        async_copy_b128(ldsKp + (uint32_t)(k * KPP_ + d4) * 4u,
                        KpG + (size_t)k * HEAD_DIM_ + d4);
    }
    const float* QpG = Qp + (size_t)bh * (L_ * HEAD_DIM_) + (size_t)q0 * HEAD_DIM_;
    for (int i = tid; i < QT_ * (HEAD_DIM_ / 4); i += 256) {
        int q = i >> 4, d4 = (i & 15) * 4;
        async_copy_b128(ldsQp + (uint32_t)(q * KPP_ + d4) * 4u,
                        QpG + (size_t)q * HEAD_DIM_ + d4);
    }
    if (tid < HEAD_DIM_) vS[tid] = vvec[tid];
    wait_async_all();          // our wave's async LDS writes landed
    __syncthreads();           // everyone's landed

    // ---- additive scores: s[q,k] = sum_d v[d]*tanh(Kp[k,d]+Qp[q,d]) ----
    // lanes sweep consecutive k -> Kp banks (k+d)%64 conflict-free
    for (int p = tid; p < QT_ * L_; p += 256) {
        int q = p >> 9, k = p & (L_ - 1);
        const float* kr = KpS + k * KPP_;
        const float* qr = QpS + q * KPP_;
        float acc = 0.0f;
#pragma unroll 8
        for (int d = 0; d < HEAD_DIM_; ++d)
            acc += vS[d] * fast_tanh(kr[d] + qr[d]);
        scS[q * SCP_ + k] = acc;
    }
    __syncthreads();           // Kp region now dead; scS fully written

    // ---- async-stage V into the Kp LDS region (overlaps with softmax) ----
    const float* VG = value + (size_t)b * L_ * HID_ + h * HEAD_DIM_;
    for (int i = tid; i < L_ * (HEAD_DIM_ / 4); i += 256) {
        int k = i >> 4, d4 = (i & 15) * 4;
        async_copy_b128(ldsKp + (uint32_t)(k * KPP_ + d4) * 4u,
                        VG + (size_t)k * HID_ + d4);
    }

    // ---- softmax over k (each wave owns 4 rows), write attn to d_out ----
    // attn is written once and never re-read on-device: stream it (NT stores).
    float* attnG = attn_out + ((size_t)bh * L_ + q0) * L_;
    for (int i = 0; i < 4; ++i) {
        int q = wave * 4 + i;
        float* srow = scS + q * SCP_;
        float mx = -1e30f;
        for (int k = lane; k < L_; k += 32) mx = fmaxf(mx, srow[k]);
        for (int off = 16; off; off >>= 1) mx = fmaxf(mx, __shfl_xor(mx, off));
        float lsum = 0.0f;
        for (int k = lane; k < L_; k += 32) {
            float e = __expf(srow[k] - mx);
            srow[k] = e;
            lsum += e;
        }
        for (int off = 16; off; off >>= 1) lsum += __shfl_xor(lsum, off);
        float inv = __frcp_rn(lsum);
        for (int k = lane; k < L_; k += 32) {
            float pr = srow[k] * inv;
            srow[k] = pr;
            __builtin_nontemporal_store(pr, attnG + (size_t)q * L_ + k);
        }
    }
    wait_async_all();          // V tile in LDS
    __syncthreads();

    // ---- x = attn @ V via WMMA: 8 waves cover 2 m-tiles x 4 n-tiles ----
    const int mt = wave >> 2, nt = wave & 3;
    const int hi = lane >> 4, ln = lane & 15;
    const float* arow = scS + (size_t)(mt * 16 + ln) * SCP_;   // 8B aligned rows
    v8f c = {};
    for (int kk = 0; kk < L_; kk += 4) {
        v2f a  = *(const v2f*)(arow + kk + 2 * hi);            // one ds_load_b64
        v2f bb;
        bb.x = KpS[(kk + 2 * hi) * KPP_ + nt * 16 + ln];       // V in LDS
        bb.y = KpS[(kk + 2 * hi + 1) * KPP_ + nt * 16 + ln];
        c = __builtin_amdgcn_wmma_f32_16x16x4_f32(false, a, false, bb,
                                                  (short)0, c, false, false);
    }
    // x stored as [B, L, H*64] (matches reference transpose+reshape)
    float* xG = x_out + ((size_t)b * L_ + q0 + mt * 16) * HID_
                      + h * HEAD_DIM_ + nt * 16 + ln;
#pragma unroll
    for (int r = 0; r < 8; ++r)
        xG[(size_t)(r + 8 * hi) * HID_] = c[r];
}

// ---------------------------------------------------------------------------
// Kernel 3: out = x @ Wo^T + bo.  M=2048, N=512, K=512; one wave per tile.
// ---------------------------------------------------------------------------
__global__ __launch_bounds__(256) void out_kernel(
    const float* __restrict__ x, const float* __restrict__ Wo,
    const float* __restrict__ bo, float* __restrict__ out)
{
    const int lane = threadIdx.x & 31, wave = threadIdx.x >> 5;
    const int tile = blockIdx.x * 8 + wave;     // 0..4095
    const int mt = tile & 127;                  // 128 row tiles
    const int nt = tile >> 7;                   // 32 col tiles
    const int hi = lane >> 4, ln = lane & 15;

    const float* arow = x + (size_t)(mt * 16 + ln) * HID_;
    const float* brow = Wo + (size_t)(nt * 16 + ln) * HID_;   // B[k][n]=Wo[n][k]

    v8f c = {};
    for (int kk = 0; kk < HID_; kk += 4) {
        v2f a, bb;
        a.x  = arow[kk + 2 * hi];
        a.y  = arow[kk + 2 * hi + 1];
        bb.x = brow[kk + 2 * hi];
        bb.y = brow[kk + 2 * hi + 1];
        c = __builtin_amdgcn_wmma_f32_16x16x4_f32(false, a, false, bb,
                                                  (short)0, c, false, false);
    }
#pragma unroll
    for (int r = 0; r < 8; ++r) {
        int row = mt * 16 + r + 8 * hi;
        int col = nt * 16 + ln;
        out[(size_t)row * HID_ + col] = c[r] + bo[col];
    }
}

// ---------------------------------------------------------------------------
extern "C" void kernel_launch(void* const* d_in, const int* in_sizes, int n_in,
                              void* d_out, int out_size, void* d_ws, size_t ws_size,
                              hipStream_t stream) {
    const float* query = (const float*)d_in[0];
    const float* key   = (const float*)d_in[1];
    const float* value = (const float*)d_in[2];
    const float* W1    = (const float*)d_in[3];
    const float* W2    = (const float*)d_in[4];
    const float* v     = (const float*)d_in[5];
    const float* Wo    = (const float*)d_in[6];
    const float* bo    = (const float*)d_in[7];

    float* out  = (float*)d_out;                          // [4,512,512]
    float* attn = out + (size_t)B_ * L_ * HID_;           // [4,8,512,512]

    float* ws = (float*)d_ws;
    const size_t per = (size_t)B_ * N_HEADS_ * L_ * HEAD_DIM_;  // 1,048,576
    float* Qp = ws;
    float* Kp = Qp + per;
    float* x  = Kp + per;                                 // [4,512,512]

    // 1) projections: grid (16 tile-groups, 32 heads, {Q,K})
    proj_kernel<<<dim3(16, 32, 2), 256, 0, stream>>>(query, key, W2, W1, Qp, Kp);

    // 2) scores + softmax + attn@V: dynamic LDS ~208 KB (<= 320 KB WGP LDS)
    size_t smem = (size_t)(L_ * KPP_ + QT_ * SCP_ + QT_ * KPP_ + 64) * sizeof(float);
    attn_kernel<<<dim3(L_ / QT_, B_ * N_HEADS_), 256, smem, stream>>>(
        Qp, Kp, value, v, attn, x);

    // 3) output projection: 4096 tiles / 8 waves per block
    out_kernel<<<dim3(512), 256, 0, stream>>>(x, Wo, bo, out);
}